// MSELoss_84817014161982
// MI455X (gfx1250) — compile-verified
//
#include <hip/hip_runtime.h>
#include <hip/hip_bf16.h>
#include <math.h>

typedef __attribute__((ext_vector_type(2))) float v2f;
typedef __attribute__((ext_vector_type(8))) float v8f;
typedef __attribute__((ext_vector_type(4))) unsigned int v4u;
typedef __attribute__((ext_vector_type(8))) int v8i;
typedef __attribute__((ext_vector_type(4))) int v4i;

#define TILE 256
#define CHUNKS 4
#define PSTRIDE 20
#define WEIGHT_MSE (1.0f / 3.0f)

// ---------------------------------------------------------------------------
// Kernel 1: per-batch precompute.  stats[b] = {maskSum, Stt}, and the per-atom
// weight w = (1+5*dna+5*rna+10*lig)*mask into wglob[b*N+n] so the streaming
// kernel does 1 float load instead of 4 int loads per atom.
// ---------------------------------------------------------------------------
__global__ void __launch_bounds__(256) k1_batch_stats(
    const float* __restrict__ true_c, const int* __restrict__ mask,
    const int* __restrict__ dna, const int* __restrict__ rna,
    const int* __restrict__ lig, float* __restrict__ stats,
    float* __restrict__ wglob, int N) {
  int b = blockIdx.x, tid = threadIdx.x;
  float ms = 0.f, stt = 0.f;
  for (int n = tid; n < N; n += 256) {
    size_t gi = (size_t)b * N + n;
    float m = (float)mask[gi];
    float w = (1.0f + 5.0f * (float)dna[gi] + 5.0f * (float)rna[gi] +
               10.0f * (float)lig[gi]) * m;
    wglob[gi] = w;
    float tx = true_c[gi * 3 + 0];
    float ty = true_c[gi * 3 + 1];
    float tz = true_c[gi * 3 + 2];
    ms += m;
    stt += w * (tx * tx + ty * ty + tz * tz);
  }
  __shared__ float red0[256], red1[256];
  red0[tid] = ms; red1[tid] = stt;
  __syncthreads();
  for (int off = 128; off > 0; off >>= 1) {
    if (tid < off) { red0[tid] += red0[tid + off]; red1[tid] += red1[tid + off]; }
    __syncthreads();
  }
  if (tid == 0) { stats[b * 4 + 0] = red0[0]; stats[b * 4 + 1] = red1[0]; }
}

// ---------------------------------------------------------------------------
// TDM: 1-D tile DMA of f32 elements from gaddr into LDS at byte offset
// lds_byte (OOB reads beyond tensor_dim0 return zero).  D# per ISA 8.3/8.4:
//  group0: count=1 | lds_addr | global_addr[56:0] | type=2
//  group1: data_size=4B, tensor_dim0, tensor_dim1=1, tile_dim0, tile_dim1=1
//  groups 2/3 (and the 5th SGPR group of the 6-arg builtin): zero (<=2D)
// ---------------------------------------------------------------------------
__device__ __forceinline__ void tdm_load_f32_row(unsigned lds_byte,
                                                 unsigned long long gaddr,
                                                 unsigned tensor_elems,
                                                 unsigned tile_elems) {
  v4u g0;
  g0.x = 1u;                                              // count=1, user mode
  g0.y = lds_byte;                                        // lds_addr
  g0.z = (unsigned)(gaddr & 0xFFFFFFFFu);                 // global_addr[31:0]
  g0.w = (unsigned)((gaddr >> 32) & 0x1FFFFFFu) | (2u << 30);  // [56:32]|type=2
  v8i g1;
  g1[0] = (int)(2u << 16);                                // data_size = 4 bytes
  g1[1] = (int)((tensor_elems & 0xFFFFu) << 16);          // tensor_dim0[15:0]
  g1[2] = (int)(((tensor_elems >> 16) & 0xFFFFu) | (1u << 16)); // td0 hi | td1=1
  g1[3] = (int)((tile_elems & 0xFFFFu) << 16);            // tile_dim0
  g1[4] = 1;                                              // tile_dim1 = 1
  g1[5] = (int)tensor_elems;                              // dim0 stride (1 row)
  g1[6] = 0;
  g1[7] = 0;
  v4i gz4 = {0, 0, 0, 0};
  v8i gz8 = {0, 0, 0, 0, 0, 0, 0, 0};
  __builtin_amdgcn_tensor_load_to_lds(g0, g1, gz4, gz4, gz8, 0);
}

// ---------------------------------------------------------------------------
// Kernel 2: grid = B*S*CHUNKS.  Each block streams its chunk of pred once and
// accumulates all Kabsch moments with V_WMMA_F32_16X16X4_F32:
//   A row (atom k) = (w*px, w*py, w*pz, w, w*|p|^2)   B col = (tx, ty, tz, 1)
//   C[0..2][0..2]=S_pt  C[0..2][3]=Sp  C[3][0..2]=St  C[3][3]=W  C[4][3]=Spp
// The per-batch true tile is DMA'd into LDS by the Tensor Data Mover (wave 0
// issues it, overlaps with pred staging, s_wait_tensorcnt before the barrier),
// keeping the vector-memory pipe dedicated to the 37.7MB pred stream.
// Partial moments (20 floats) go to ws for a deterministic combine.
// ---------------------------------------------------------------------------
__global__ void __launch_bounds__(256) k2_kabsch_moments(
    const float* __restrict__ pred, const float* __restrict__ true_c,
    const float* __restrict__ wglob, float* __restrict__ partials,
    int S, int N) {
  int x = blockIdx.x;
  int bs = x / CHUNKS;
  int ch = x % CHUNKS;
  int b = bs / S;
  int tid = threadIdx.x;
  int wave = tid >> 5;
  int lane = tid & 31;
  int half = lane >> 4;
  int idx = lane & 15;

  __shared__ float ldsTrue[TILE * 3];   // raw true tile (TDM destination)
  __shared__ float ap[TILE * 5];        // augmented weighted pred
  __shared__ float waveC[8][5][16];
  __shared__ float accC[5 * 16];

  const float* predBS = pred + (size_t)bs * N * 3;
  const float* trueB  = true_c + (size_t)b * N * 3;
  const float* wB     = wglob + (size_t)b * N;

  int per = (N + CHUNKS - 1) / CHUNKS;
  int start = ch * per;
  int end = min(N, start + per);

  unsigned ldsTrueOff = (unsigned)(uintptr_t)(void*)ldsTrue;  // flat low 32 = LDS offset

  v8f c = {};

  for (int base = start; base < end; base += TILE) {
    // Kick the TDM for this tile's true data (wave 0 only; EXEC-independent,
    // one DMA per block).  tensor_dim0 clamps the tail: OOB reads -> 0.
    if (wave == 0) {
      tdm_load_f32_row(ldsTrueOff,
                       (unsigned long long)(uintptr_t)(trueB + (size_t)base * 3),
                       (unsigned)((N - base) * 3), TILE * 3);
    }

    // Stage augmented weighted pred while the DMA is in flight.
    int n = base + tid;
    float w = 0.f, px = 0.f, py = 0.f, pz = 0.f;
    if (n < end) {
      w = wB[n];
      px = predBS[(size_t)n * 3 + 0];
      py = predBS[(size_t)n * 3 + 1];
      pz = predBS[(size_t)n * 3 + 2];
    }
    ap[tid * 5 + 0] = w * px;
    ap[tid * 5 + 1] = w * py;
    ap[tid * 5 + 2] = w * pz;
    ap[tid * 5 + 3] = w;
    ap[tid * 5 + 4] = w * (px * px + py * py + pz * pz);

    if (wave == 0) __builtin_amdgcn_s_wait_tensorcnt(0);
    __syncthreads();

    int sub = wave * 32;
#pragma unroll
    for (int g = 0; g < 8; ++g) {
      int a0 = sub + g * 4 + half * 2;  // K = half*2 (+ vgpr index)
      v2f A, Bv;
      A.x  = (idx < 5) ? ap[a0 * 5 + idx] : 0.f;
      A.y  = (idx < 5) ? ap[(a0 + 1) * 5 + idx] : 0.f;
      Bv.x = (idx < 3) ? ldsTrue[a0 * 3 + idx] : ((idx == 3) ? 1.f : 0.f);
      Bv.y = (idx < 3) ? ldsTrue[(a0 + 1) * 3 + idx] : ((idx == 3) ? 1.f : 0.f);
      c = __builtin_amdgcn_wmma_f32_16x16x4_f32(
          false, A, false, Bv, (short)0, c, false, false);
    }
    __syncthreads();
  }

  // C layout: VGPR r, lane l -> row M = r + (l<16?0:8), col = l%16.
  // Rows 0..4 live entirely in lanes 0..15.  Fixed-order cross-wave sum.
  if (lane < 16) {
#pragma unroll
    for (int r = 0; r < 5; ++r) waveC[wave][r][idx] = c[r];
  }
  __syncthreads();
  if (tid < 80) {
    float a = 0.f;
    int r = tid >> 4, col = tid & 15;
#pragma unroll
    for (int wv = 0; wv < 8; ++wv) a += waveC[wv][r][col];
    accC[tid] = a;
  }
  __syncthreads();

  // Emit rows 0..4, cols 0..3 (20 floats; 17 used downstream).
  if (tid < PSTRIDE)
    partials[(size_t)x * PSTRIDE + tid] = accC[(tid >> 2) * 16 + (tid & 3)];
}

// ---------------------------------------------------------------------------
// Kernel 3: deterministic finalize.  One thread per (b,s): combine chunk
// partials in fixed order, closed-form 3x3 singular values of H, RMSD
// identity, then fixed-order block reduction to the scalar loss.
// ---------------------------------------------------------------------------
__global__ void __launch_bounds__(256) k3_finalize(
    const float* __restrict__ partials, const float* __restrict__ stats,
    const float* __restrict__ scale, float* __restrict__ out, int B, int S) {
  int tid = threadIdx.x;
  int count = B * S;
  float acc = 0.f;

  for (int bs = tid; bs < count; bs += 256) {
    int b = bs / S;
    float P[PSTRIDE];
#pragma unroll
    for (int i = 0; i < PSTRIDE; ++i) P[i] = 0.f;
    for (int chv = 0; chv < CHUNKS; ++chv) {
      const float* src = partials + (size_t)(bs * CHUNKS + chv) * PSTRIDE;
#pragma unroll
      for (int i = 0; i < PSTRIDE; ++i) P[i] += src[i];
    }
    // Decode: row r, col c at P[r*4+c]
    float Spt[3][3], Sp[3], St[3];
#pragma unroll
    for (int i = 0; i < 3; ++i) {
#pragma unroll
      for (int j = 0; j < 3; ++j) Spt[i][j] = P[i * 4 + j];
      Sp[i] = P[i * 4 + 3];
      St[i] = P[12 + i];
    }
    float W = P[15], Spp = P[19];
    float maskSum = stats[b * 4 + 0];
    float Stt = stats[b * 4 + 1];

    float ps = 0.f;
    if (W > 0.f) {
      float invW = 1.0f / W;
      float H[3][3];
#pragma unroll
      for (int i = 0; i < 3; ++i)
#pragma unroll
        for (int j = 0; j < 3; ++j)
          H[i][j] = Spt[i][j] - Sp[i] * St[j] * invW;

      float detH = H[0][0] * (H[1][1] * H[2][2] - H[1][2] * H[2][1])
                 - H[0][1] * (H[1][0] * H[2][2] - H[1][2] * H[2][0])
                 + H[0][2] * (H[1][0] * H[2][1] - H[1][1] * H[2][0]);

      float K00 = H[0][0]*H[0][0] + H[1][0]*H[1][0] + H[2][0]*H[2][0];
      float K11 = H[0][1]*H[0][1] + H[1][1]*H[1][1] + H[2][1]*H[2][1];
      float K22 = H[0][2]*H[0][2] + H[1][2]*H[1][2] + H[2][2]*H[2][2];
      float K01 = H[0][0]*H[0][1] + H[1][0]*H[1][1] + H[2][0]*H[2][1];
      float K02 = H[0][0]*H[0][2] + H[1][0]*H[1][2] + H[2][0]*H[2][2];
      float K12 = H[0][1]*H[0][2] + H[1][1]*H[1][2] + H[2][1]*H[2][2];

      float q  = (K00 + K11 + K22) * (1.0f / 3.0f);
      float p1 = K01 * K01 + K02 * K02 + K12 * K12;
      float d0 = K00 - q, d1 = K11 - q, d2 = K22 - q;
      float p2 = d0 * d0 + d1 * d1 + d2 * d2 + 2.0f * p1;

      float s1, s2, s3;
      if (p2 <= 1e-12f * q * q + 1e-30f) {
        s1 = s2 = s3 = sqrtf(fmaxf(q, 0.f));
      } else {
        float p  = sqrtf(p2 * (1.0f / 6.0f));
        float ip = 1.0f / p;
        float B00 = d0 * ip, B11 = d1 * ip, B22 = d2 * ip;
        float B01 = K01 * ip, B02 = K02 * ip, B12 = K12 * ip;
        float detB = B00 * (B11 * B22 - B12 * B12)
                   - B01 * (B01 * B22 - B12 * B02)
                   + B02 * (B01 * B12 - B11 * B02);
        float r = fminf(fmaxf(0.5f * detB, -1.0f), 1.0f);
        float phi = acosf(r) * (1.0f / 3.0f);
        float l1 = q + 2.0f * p * cosf(phi);
        float l3 = q + 2.0f * p * cosf(phi + 2.0943951023931953f);
        float l2 = 3.0f * q - l1 - l3;
        s1 = sqrtf(fmaxf(l1, 0.f));
        s2 = sqrtf(fmaxf(l2, 0.f));
        s3 = sqrtf(fmaxf(l3, 0.f));
      }
      float sumS = s1 + s2 + ((detH < 0.f) ? -s3 : s3);
      float nSp = Sp[0]*Sp[0] + Sp[1]*Sp[1] + Sp[2]*Sp[2];
      float nSt = St[0]*St[0] + St[1]*St[1] + St[2]*St[2];
      float swse = Spp + Stt - (nSp + nSt) * invW - 2.0f * sumS;
      ps = swse / (maskSum + 1e-6f) * scale[bs];
    }
    acc += ps;
  }

  __shared__ float red[256];
  red[tid] = acc;
  __syncthreads();
  for (int off = 128; off > 0; off >>= 1) {
    if (tid < off) red[tid] += red[tid + off];
    __syncthreads();
  }
  if (tid == 0) out[0] = red[0] * (WEIGHT_MSE / (float)count);
}

// ---------------------------------------------------------------------------
extern "C" void kernel_launch(void* const* d_in, const int* in_sizes, int n_in,
                              void* d_out, int out_size, void* d_ws,
                              size_t ws_size, hipStream_t stream) {
  const float* pred  = (const float*)d_in[0];
  const float* truec = (const float*)d_in[1];
  const int*   mask  = (const int*)d_in[2];
  const int*   dna   = (const int*)d_in[3];
  const int*   rna   = (const int*)d_in[4];
  const int*   lig   = (const int*)d_in[5];
  const float* scale = (const float*)d_in[6];

  int S = in_sizes[0] / in_sizes[1];   // pred / true = S
  int B = in_sizes[6] / S;             // scale = B*S
  int N = in_sizes[1] / (3 * B);

  float* stats    = (float*)d_ws;                 // B*4 floats
  float* wglob    = stats + 4 * B;                // B*N floats
  float* partials = wglob + (size_t)B * N;        // B*S*CHUNKS*PSTRIDE floats

  k1_batch_stats<<<B, 256, 0, stream>>>(truec, mask, dna, rna, lig,
                                        stats, wglob, N);
  k2_kabsch_moments<<<B * S * CHUNKS, 256, 0, stream>>>(
      pred, truec, wglob, partials, S, N);
  k3_finalize<<<1, 256, 0, stream>>>(partials, stats, scale,
                                     (float*)d_out, B, S);
}